// LSTMPredictV2_18296560681571
// MI455X (gfx1250) — compile-verified
//
#include <hip/hip_runtime.h>
#include <math.h>

typedef __attribute__((ext_vector_type(2))) float v2f;
typedef __attribute__((ext_vector_type(8))) float v8f;

#define BSZ   1024
#define TT    64
#define ISZ   1024
#define HSZ   1024
#define G4    4096
#define SEQ_T_STRIDE (TT * ISZ)   /* 65536 floats per batch row of seq */

#define GX_OFF  ((size_t)0)
#define HS_OFF  ((size_t)16 * 1024 * 1024)
#define HG_OFF  ((size_t)20 * 1024 * 1024)   /* 2 x 1024 f32 (double buffered) */
#define CNT_OFF (HG_OFF + 8192)

#define NB_SCAN  128u   /* blocks in the persistent scan */
#define NT_SCAN  512    /* threads per scan block */
#define JPB      8      /* h elements owned per block (NB_SCAN * JPB = 1024) */

__device__ __forceinline__ float sigf(float x) { return 1.0f / (1.0f + expf(-x)); }

// ---------------------------------------------------------------------------
// Init: zero both h buffers and the global barrier counter (fresh every launch).
// ---------------------------------------------------------------------------
__global__ void lstm_init_kernel(float* __restrict__ h_g, unsigned* __restrict__ counter) {
    int tid = threadIdx.x;                 // 256 threads x 8 floats = 2048
    float4 z = make_float4(0.f, 0.f, 0.f, 0.f);
    ((float4*)h_g)[tid * 2]     = z;
    ((float4*)h_g)[tid * 2 + 1] = z;
    if (tid == 0) *counter = 0u;
}

// ---------------------------------------------------------------------------
// gx = x @ W_ih^T + (b_ih + b_hh), x = seq[:, -1, :]
// fp32 WMMA 16x16x4. Block = 8 waves covering a 32(M) x 64(N) tile.
// ---------------------------------------------------------------------------
__global__ void gx_gemm_kernel(const float* __restrict__ seq,
                               const float* __restrict__ Wih,
                               const float* __restrict__ bih,
                               const float* __restrict__ bhh,
                               float* __restrict__ gx) {
    const int lane  = threadIdx.x & 31;
    const int wave  = threadIdx.x >> 5;      // 0..7
    const int waveM = wave >> 2;             // 0..1
    const int waveN = wave & 3;              // 0..3

    const int bm = blockIdx.x >> 6;          // 0..31  (1024/32)
    const int bn = blockIdx.x & 63;          // 0..63  (4096/64)

    const int m0 = bm * 32 + waveM * 16;
    const int n0 = bn * 64 + waveN * 16;

    const int l16   = lane & 15;
    const int khalf = (lane >> 4) * 2;       // VGPR pair covers K = khalf, khalf+1

    // A: row m = m0 + l16 of x = seq[m, 63, :]
    const float* arow = seq + (size_t)(m0 + l16) * SEQ_T_STRIDE + (size_t)(TT - 1) * ISZ;
    // B: row n = n0 + l16 of W_ih (row-major 4096 x 1024, K along the row)
    const float* brow = Wih + (size_t)(n0 + l16) * ISZ;

    v8f acc = {};
    #pragma unroll 4
    for (int k = 0; k < ISZ; k += 4) {
        v2f a = *(const v2f*)(arow + k + khalf);
        v2f b = *(const v2f*)(brow + k + khalf);
        // (neg_a, A, neg_b, B, c_mod, C, reuse_a, reuse_b)
        acc = __builtin_amdgcn_wmma_f32_16x16x4_f32(false, a, false, b,
                                                    (short)0, acc, false, false);
    }

    const int   n    = n0 + l16;
    const float bias = bih[n] + bhh[n];
    const int   mb   = m0 + ((lane >> 4) ? 8 : 0);   // C layout: rows i / i+8 per half-wave
    #pragma unroll
    for (int i = 0; i < 8; ++i) {
        gx[(size_t)(mb + i) * G4 + n] = acc[i] + bias;
    }
}

// ---------------------------------------------------------------------------
// Sequential scan over 1024 steps. Persistent grid of 128 blocks x 512 thr.
// Block b owns h indices [8b, 8b+8): computes rows {g*1024 + 8b + j} of
// W_hh @ h for g in 0..3, applies gates locally, publishes h via L2.
// h is double-buffered (read buf[t&1], write buf[(t+1)&1]) so a single
// monotonic-counter barrier per step is race-free even if blocks drift a
// full step apart.
// ---------------------------------------------------------------------------
__global__ void lstm_scan_kernel(const float* __restrict__ gx,
                                 const float* __restrict__ Whh,
                                 float* __restrict__ hs,
                                 float* __restrict__ h_g,     // 2 x 1024 floats
                                 unsigned* __restrict__ counter) {
    __shared__ float hsh[HSZ];
    __shared__ float gsum[32];
    __shared__ float c_s[JPB];

    const int tid = threadIdx.x;
    const int r   = tid >> 4;       // 0..31 : row group = (gate g = r>>3, joff = r&7)
    const int li  = tid & 15;       // lane within row group

    const int  jbase = blockIdx.x * JPB;
    const int  row   = (r >> 3) * HSZ + jbase + (r & 7);
    const float* wrow = Whh + (size_t)row * HSZ;

    if (tid < JPB) c_s[tid] = 0.0f;

    for (int t = 0; t < BSZ; ++t) {
        const float* hrd = h_g + (t & 1) * HSZ;
        float*       hwr = h_g + ((t + 1) & 1) * HSZ;

        // Stage current h into LDS (agent-scope loads: coherent view of L2).
        #pragma unroll
        for (int q = 0; q < 2; ++q) {
            int idx = tid * 2 + q;
            hsh[idx] = __hip_atomic_load(&hrd[idx], __ATOMIC_RELAXED,
                                         __HIP_MEMORY_SCOPE_AGENT);
        }
        __syncthreads();

        // Partial dot: 64 FMAs per thread, W_hh streamed from L2-resident set.
        float s = 0.0f;
        #pragma unroll 8
        for (int i = 0; i < 64; ++i) {
            int k = li + 16 * i;
            s = fmaf(wrow[k], hsh[k], s);
        }
        // Reduce within the 16-lane row group (stays inside wave32 halves).
        s += __shfl_xor(s, 8, 32);
        s += __shfl_xor(s, 4, 32);
        s += __shfl_xor(s, 2, 32);
        s += __shfl_xor(s, 1, 32);
        if (li == 0) gsum[r] = s + gx[(size_t)t * G4 + row];
        __syncthreads();

        // Gate math + state update for the JPB owned h elements.
        if (tid < JPB) {
            float iv = gsum[tid];
            float fv = gsum[JPB + tid];
            float gv = gsum[2 * JPB + tid];
            float ov = gsum[3 * JPB + tid];
            float c  = sigf(fv) * c_s[tid] + sigf(iv) * tanhf(gv);
            float h  = sigf(ov) * tanhf(c);
            c_s[tid] = c;
            int j = jbase + tid;
            __hip_atomic_store(&hwr[j], h, __ATOMIC_RELAXED, __HIP_MEMORY_SCOPE_AGENT);
            hs[(size_t)t * HSZ + j] = h;
        }
        __threadfence();
        __syncthreads();

        // Grid barrier: monotonic counter (128 * 1024 < 2^31, no wraparound).
        if (tid == 0) {
            __hip_atomic_fetch_add(counter, 1u, __ATOMIC_ACQ_REL,
                                   __HIP_MEMORY_SCOPE_AGENT);
            unsigned target = NB_SCAN * (unsigned)(t + 1);
            while (__hip_atomic_load(counter, __ATOMIC_ACQUIRE,
                                     __HIP_MEMORY_SCOPE_AGENT) < target) {
                __builtin_amdgcn_s_sleep(1);
            }
        }
        __syncthreads();
    }
}

// ---------------------------------------------------------------------------
// out[t] = dot(hs[t], W_reg[0]) + b_reg[0]
// ---------------------------------------------------------------------------
__global__ void head_kernel(const float* __restrict__ hs,
                            const float* __restrict__ Wreg,
                            const float* __restrict__ breg,
                            float* __restrict__ out) {
    __shared__ float red[8];
    const int t   = blockIdx.x;
    const int tid = threadIdx.x;

    float4 a = ((const float4*)(hs + (size_t)t * HSZ))[tid];
    float4 b = ((const float4*)Wreg)[tid];
    float  s = a.x * b.x + a.y * b.y + a.z * b.z + a.w * b.w;

    s += __shfl_xor(s, 16, 32);
    s += __shfl_xor(s, 8, 32);
    s += __shfl_xor(s, 4, 32);
    s += __shfl_xor(s, 2, 32);
    s += __shfl_xor(s, 1, 32);
    if ((tid & 31) == 0) red[tid >> 5] = s;
    __syncthreads();
    if (tid == 0) {
        float tot = 0.0f;
        #pragma unroll
        for (int w = 0; w < 8; ++w) tot += red[w];
        out[t] = tot + breg[0];
    }
}

// ---------------------------------------------------------------------------
extern "C" void kernel_launch(void* const* d_in, const int* in_sizes, int n_in,
                              void* d_out, int out_size, void* d_ws, size_t ws_size,
                              hipStream_t stream) {
    (void)in_sizes; (void)n_in; (void)out_size; (void)ws_size;

    const float* seq  = (const float*)d_in[0];
    const float* Wih  = (const float*)d_in[1];
    const float* Whh  = (const float*)d_in[2];
    const float* bih  = (const float*)d_in[3];
    const float* bhh  = (const float*)d_in[4];
    const float* Wreg = (const float*)d_in[5];
    const float* breg = (const float*)d_in[6];
    float* out = (float*)d_out;

    char* ws = (char*)d_ws;
    float*    gx  = (float*)(ws + GX_OFF);     // 1024 x 4096 f32 (16 MB)
    float*    hs  = (float*)(ws + HS_OFF);     // 1024 x 1024 f32 ( 4 MB)
    float*    hg  = (float*)(ws + HG_OFF);     // 2 x 1024 f32 (double-buffered h)
    unsigned* cnt = (unsigned*)(ws + CNT_OFF); // barrier counter

    lstm_init_kernel<<<1, 256, 0, stream>>>(hg, cnt);
    gx_gemm_kernel<<<2048, 256, 0, stream>>>(seq, Wih, bih, bhh, gx);
    lstm_scan_kernel<<<NB_SCAN, NT_SCAN, 0, stream>>>(gx, Whh, hs, hg, cnt);
    head_kernel<<<1024, 256, 0, stream>>>(hs, Wreg, breg, out);
}